// ResConvBlock_56676388438705
// MI455X (gfx1250) — compile-verified
//
#include <hip/hip_runtime.h>

#define N_PTS 100000
#define COUT  32
#define KVOL  27
#define BN_EPS 1e-5f
#define LRELU_SLOPE 0.01f
#define PTS_PER_BLOCK 128   // 256 threads = 8 waves x 16 points

typedef __attribute__((ext_vector_type(2))) float v2f;
typedef __attribute__((ext_vector_type(8))) float v8f;

__device__ __forceinline__ float lrelu(float x) { return x >= 0.0f ? x : LRELU_SLOPE * x; }

// ---------------------------------------------------------------------------
// Zero BN-stat accumulators (ws is poisoned; re-zero every launch).
// layout: sum1[32] sq1[32] sum2[32] sq2[32] sumr[32] sqr[32]
// ---------------------------------------------------------------------------
__global__ __launch_bounds__(256) void rcb_zero_stats(float* __restrict__ stats) {
  int t = threadIdx.x;
  if (t < 6 * COUT) stats[t] = 0.0f;
}

// ---------------------------------------------------------------------------
// Gather-GEMM submanifold conv: Y[n,:] = sum_k mask[n,k] * X[nbr[n,k],:] @ W[k]
// Wave = 16 points, two 16x16 tiles via V_WMMA_F32_16X16X4_F32.
// Software pipelined: neighbor idx/mask fetched 2 iterations ahead, gathered
// A row 1 iteration ahead; W k-slice double-buffered in LDS (1 barrier / k).
// Mask applied as a multiply (no divergence). Batch stats via atomics.
// RESID: additionally R = X @ Wres (conv1 path, CIN==16 only).
// ---------------------------------------------------------------------------
template<int CIN, bool RESID>
__global__ __launch_bounds__(256) void rcb_subm_conv(
    const float* __restrict__ X,              // [N, CIN]
    const int* __restrict__ nbr,              // [N, KVOL]
    const unsigned char* __restrict__ msk,    // [N, KVOL] (jnp bool = 1 byte)
    const float* __restrict__ W,              // [KVOL, CIN, 32]
    float* __restrict__ Y,                    // [N, 32] pre-BN output
    float* __restrict__ sumY, float* __restrict__ sqY,
    const float* __restrict__ Wres,           // [CIN, 32] (RESID only)
    float* __restrict__ R,
    float* __restrict__ sumR, float* __restrict__ sqR)
{
  constexpr int NSTEP  = CIN / 4;
  constexpr int WSLICE = CIN * COUT;
  __shared__ __align__(16) float sW[2][WSLICE];

  const int tid  = threadIdx.x;
  const int lane = tid & 31;
  const int wave = tid >> 5;
  const int m16  = lane & 15;                 // A row / output-channel lane
  const int half = lane >> 4;                 // K-half of the 4-deep fragment
  const int chi  = 2 * half;
  const int n0   = blockIdx.x * PTS_PER_BLOCK + wave * 16;
  const int na   = min(n0 + m16, N_PTS - 1);
  const bool full = (n0 + 16) <= N_PTS;

  // stage W slice k=0 into buffer 0 (float4)
  {
    const float4* src = (const float4*)W;
    float4* dst = (float4*)sW[0];
    for (int i = tid; i < WSLICE / 4; i += 256) dst[i] = src[i];
  }

  const int* pnb = nbr + (size_t)na * KVOL;
  const unsigned char* pmk = msk + (size_t)na * KVOL;

  // pipeline prologue: gather row k=0, prefetch idx/mask for k=1
  v2f arA[NSTEP], arB[NSTEP];
  float mvA = 0.0f, mvB = 0.0f;
  int idxN; float mvN;
  {
    int idx0 = pnb[0];
    mvA = (float)pmk[0];
    const float* px = X + (size_t)idx0 * CIN + chi;
#pragma unroll
    for (int s = 0; s < NSTEP; ++s) arA[s] = *(const v2f*)(px + 4 * s);
    idxN = pnb[1];
    mvN = (float)pmk[1];
  }

  v8f acc0 = {};  // channels 0..15
  v8f acc1 = {};  // channels 16..31

  __syncthreads();  // sW[0] staged

  auto body = [&](v2f (&cur)[NSTEP], float& mvCur, v2f (&nxt)[NSTEP], float& mvNxt, int k) {
    // issue gather of row k+1 (address regs already resident from k-1)
    if (k + 1 < KVOL) {
      const float* px = X + (size_t)idxN * CIN + chi;
      mvNxt = mvN;
#pragma unroll
      for (int s = 0; s < NSTEP; ++s) nxt[s] = *(const v2f*)(px + 4 * s);
      if (k + 2 < KVOL) { idxN = pnb[k + 2]; mvN = (float)pmk[k + 2]; }
      // stage W slice k+1 into the other LDS buffer
      const float4* src = (const float4*)(W + (size_t)(k + 1) * WSLICE);
      float4* dst = (float4*)sW[(k + 1) & 1];
      for (int i = tid; i < WSLICE / 4; i += 256) dst[i] = src[i];
    }
    // apply mask as multiply (no divergence), then WMMA over the K-slab
#pragma unroll
    for (int s = 0; s < NSTEP; ++s) { cur[s].x *= mvCur; cur[s].y *= mvCur; }
    const float* wk = sW[k & 1];
#pragma unroll
    for (int s = 0; s < NSTEP; ++s) {
      const int c = 4 * s + chi;
      v2f b0, b1;
      b0.x = wk[c * COUT + m16];        b0.y = wk[(c + 1) * COUT + m16];
      b1.x = wk[c * COUT + 16 + m16];   b1.y = wk[(c + 1) * COUT + 16 + m16];
      acc0 = __builtin_amdgcn_wmma_f32_16x16x4_f32(false, cur[s], false, b0, (short)0, acc0, false, false);
      acc1 = __builtin_amdgcn_wmma_f32_16x16x4_f32(false, cur[s], false, b1, (short)0, acc1, false, false);
    }
    __syncthreads();  // cur W buffer free for restage; next buffer staged
  };

  for (int kk = 0; kk < KVOL / 2; ++kk) {     // k = 0..25 (register rotation)
    body(arA, mvA, arB, mvB, 2 * kk);
    body(arB, mvB, arA, mvA, 2 * kk + 1);
  }
  body(arA, mvA, arB, mvB, KVOL - 1);         // k = 26

  // ---- epilogue: store tile + per-channel batch-stat atomics ----
  auto epilogue = [&](const v8f& a0, const v8f& a1v, float* __restrict__ Yp,
                      float* __restrict__ sum, float* __restrict__ sq) {
    float s0 = 0.0f, q0 = 0.0f, s1 = 0.0f, q1 = 0.0f;
    if (full) {
#pragma unroll
      for (int j = 0; j < 8; ++j) {           // lane holds rows j + 8*half
        int n = n0 + j + 8 * half;
        float v0 = a0[j], v1 = a1v[j];
        Yp[(size_t)n * COUT + m16]      = v0;
        Yp[(size_t)n * COUT + 16 + m16] = v1;
        s0 += v0; q0 += v0 * v0; s1 += v1; q1 += v1 * v1;
      }
    } else {
#pragma unroll
      for (int j = 0; j < 8; ++j) {
        int n = n0 + j + 8 * half;
        if (n < N_PTS) {
          float v0 = a0[j], v1 = a1v[j];
          Yp[(size_t)n * COUT + m16]      = v0;
          Yp[(size_t)n * COUT + 16 + m16] = v1;
          s0 += v0; q0 += v0 * v0; s1 += v1; q1 += v1 * v1;
        }
      }
    }
    s0 += __shfl_xor(s0, 16, 32); q0 += __shfl_xor(q0, 16, 32);
    s1 += __shfl_xor(s1, 16, 32); q1 += __shfl_xor(q1, 16, 32);
    if (half == 0) {
      atomicAdd(&sum[m16], s0);      atomicAdd(&sq[m16], q0);
      atomicAdd(&sum[16 + m16], s1); atomicAdd(&sq[16 + m16], q1);
    }
  };
  epilogue(acc0, acc1, Y, sumY, sqY);

  // ---- residual 1x1 GEMM: R = X @ Wres (conv1 kernel only) ----
  if constexpr (RESID) {
    // last body() barrier guarantees sW free
    {
      const float4* src = (const float4*)Wres;
      float4* dst = (float4*)sW[0];
      for (int i = tid; i < WSLICE / 4; i += 256) dst[i] = src[i];
    }
    __syncthreads();

    v8f r0 = {}, r1 = {};
    const float* px = X + (size_t)na * CIN + chi;
    const float* wk = sW[0];
#pragma unroll
    for (int s = 0; s < NSTEP; ++s) {
      const int c = 4 * s + chi;
      v2f a = *(const v2f*)(px + 4 * s);
      v2f b0, b1;
      b0.x = wk[c * COUT + m16];        b0.y = wk[(c + 1) * COUT + m16];
      b1.x = wk[c * COUT + 16 + m16];   b1.y = wk[(c + 1) * COUT + 16 + m16];
      r0 = __builtin_amdgcn_wmma_f32_16x16x4_f32(false, a, false, b0, (short)0, r0, false, false);
      r1 = __builtin_amdgcn_wmma_f32_16x16x4_f32(false, a, false, b1, (short)0, r1, false, false);
    }
    epilogue(r0, r1, R, sumR, sqR);
  }
}

// ---------------------------------------------------------------------------
// In-place a1 = lrelu(bn1(y1)) using accumulated batch stats.
// ---------------------------------------------------------------------------
__global__ __launch_bounds__(256) void rcb_bn_lrelu(
    float* __restrict__ y,
    const float* __restrict__ sum, const float* __restrict__ sq,
    const float* __restrict__ g,   const float* __restrict__ b)
{
  __shared__ float sS[COUT], sH[COUT];
  int t = threadIdx.x;
  if (t < COUT) {
    float mean = sum[t] * (1.0f / (float)N_PTS);
    float var  = sq[t] * (1.0f / (float)N_PTS) - mean * mean;
    float sc   = g[t] * rsqrtf(var + BN_EPS);
    sS[t] = sc;
    sH[t] = b[t] - mean * sc;
  }
  __syncthreads();
  size_t i = (size_t)blockIdx.x * blockDim.x + t;
  if (i < (size_t)N_PTS * COUT) {
    int c = (int)(i & (COUT - 1));
    y[i] = lrelu(fmaf(y[i], sS[c], sH[c]));
  }
}

// ---------------------------------------------------------------------------
// out = lrelu( bn2(y2) + bn_r(r) )
// ---------------------------------------------------------------------------
__global__ __launch_bounds__(256) void rcb_finalize(
    const float* __restrict__ y2, const float* __restrict__ r,
    const float* __restrict__ sum2, const float* __restrict__ sq2,
    const float* __restrict__ g2,   const float* __restrict__ b2,
    const float* __restrict__ sumr, const float* __restrict__ sqr,
    const float* __restrict__ gin,  const float* __restrict__ bin_,
    float* __restrict__ out)
{
  __shared__ float s2[COUT], h2[COUT], sr[COUT], hr[COUT];
  int t = threadIdx.x;
  if (t < COUT) {
    float m2 = sum2[t] * (1.0f / (float)N_PTS);
    float v2 = sq2[t] * (1.0f / (float)N_PTS) - m2 * m2;
    float c2 = g2[t] * rsqrtf(v2 + BN_EPS);
    s2[t] = c2; h2[t] = b2[t] - m2 * c2;

    float mr = sumr[t] * (1.0f / (float)N_PTS);
    float vr = sqr[t] * (1.0f / (float)N_PTS) - mr * mr;
    float cr = gin[t] * rsqrtf(vr + BN_EPS);
    sr[t] = cr; hr[t] = bin_[t] - mr * cr;
  }
  __syncthreads();
  size_t i = (size_t)blockIdx.x * blockDim.x + t;
  if (i < (size_t)N_PTS * COUT) {
    int c = (int)(i & (COUT - 1));
    float v = fmaf(y2[i], s2[c], h2[c]) + fmaf(r[i], sr[c], hr[c]);
    out[i] = lrelu(v);
  }
}

// ---------------------------------------------------------------------------
extern "C" void kernel_launch(void* const* d_in, const int* in_sizes, int n_in,
                              void* d_out, int out_size, void* d_ws, size_t ws_size,
                              hipStream_t stream) {
  (void)in_sizes; (void)n_in; (void)out_size; (void)ws_size;

  const float*         features  = (const float*)d_in[0];
  const int*           neighbors = (const int*)d_in[1];
  const unsigned char* mask      = (const unsigned char*)d_in[2];
  const float*         W1        = (const float*)d_in[3];
  const float*         g1        = (const float*)d_in[4];
  const float*         b1        = (const float*)d_in[5];
  const float*         W2        = (const float*)d_in[6];
  const float*         g2        = (const float*)d_in[7];
  const float*         b2        = (const float*)d_in[8];
  const float*         Win       = (const float*)d_in[9];
  const float*         gin       = (const float*)d_in[10];
  const float*         bin_      = (const float*)d_in[11];

  float* ws = (float*)d_ws;
  float* y1 = ws;                                  // [N,32]: conv1 pre-BN, then in-place a1
  float* r  = ws + (size_t)N_PTS * COUT;           // [N,32] residual pre-BN
  float* y2 = ws + 2 * (size_t)N_PTS * COUT;       // [N,32] conv2 pre-BN
  float* stats = ws + 3 * (size_t)N_PTS * COUT;    // 6*32 floats
  float* sum1 = stats;        float* sq1 = stats + 32;
  float* sum2 = stats + 64;   float* sq2 = stats + 96;
  float* sumr = stats + 128;  float* sqr = stats + 160;

  rcb_zero_stats<<<1, 256, 0, stream>>>(stats);

  dim3 gridConv((N_PTS + PTS_PER_BLOCK - 1) / PTS_PER_BLOCK);

  // conv1 (Cin=16) + residual GEMM
  rcb_subm_conv<16, true><<<gridConv, 256, 0, stream>>>(
      features, neighbors, mask, W1,
      y1, sum1, sq1,
      Win, r, sumr, sqr);

  // a1 = lrelu(bn1(y1)), in place
  size_t total = (size_t)N_PTS * COUT;
  rcb_bn_lrelu<<<(unsigned)((total + 255) / 256), 256, 0, stream>>>(y1, sum1, sq1, g1, b1);

  // conv2 (Cin=32) on activated features
  rcb_subm_conv<32, false><<<gridConv, 256, 0, stream>>>(
      y1, neighbors, mask, W2,
      y2, sum2, sq2,
      nullptr, nullptr, nullptr, nullptr);

  // out = lrelu(bn2(y2) + bn_r(r))
  rcb_finalize<<<(unsigned)((total + 255) / 256), 256, 0, stream>>>(
      y2, r, sum2, sq2, g2, b2, sumr, sqr, gin, bin_, (float*)d_out);
}